// Decoder_90108413870705
// MI455X (gfx1250) — compile-verified
//
#include <hip/hip_runtime.h>
#include <hip/hip_bf16.h>

// ---------------- problem constants (match reference) ----------------
#define VOCAB_  50000
#define OOV_    1000
#define VT_     51000          // VOCAB + OOV
#define EMB_    256
#define H_      512
#define ENC_    512
#define Bz_     32
#define S_      256
#define T_      8
#define NEG_    (-1e20f)

typedef long long ll;
typedef __attribute__((ext_vector_type(2))) float v2f;
typedef __attribute__((ext_vector_type(8))) float v8f;

// Explicit global-address-space pointers: keeps fragment loads on the
// GLOBAL_LOAD path (LOADcnt only) even when the pointers become loop PHIs.
// Without this, clang's addrspace inference fell back to FLAT loads, which
// consume LDS bandwidth and both LOADcnt+DScnt on CDNA5.
typedef const __attribute__((address_space(1))) float* gcfp;
typedef       __attribute__((address_space(1))) float* gfp;

// ---------------------------------------------------------------------
// Generic batched fp32 WMMA GEMM: C[b] = act( beta*C[b] + A[b] @ B[b] + bias )
//   A: row-major, element A[r][k] = Ab[r*lda + k]
//   B: arbitrary strides,  B[k][n] = Bb[k*ldb_k + n*ldb_n]
//   One wave computes a (16*MT) x (16*NT) block of C via
//   V_WMMA_F32_16X16X4_F32: MT A-fragments x NT B-fragments feed MT*NT
//   WMMAs per K-step (K-step = 4). Fragment loads for step kk+4 are issued
//   before the WMMA burst of step kk (software pipeline, last step peeled so
//   the steady-state loop is branch-free); all addressing is pointer
//   increment with loop-invariant strides in address space 1.
//   KS1: compile-time "ldb_k == 1" (A @ B^T case) so the B K-pair merges
//   into one b64 load. K must be a multiple of 4 (true for all call sites).
// ---------------------------------------------------------------------
template<int MT, int NT, bool KS1>
__global__ void wmma_gemm_f32(const float* __restrict__ A,
                              const float* __restrict__ Bm,
                              float* __restrict__ C,
                              const float* __restrict__ bias,
                              int M, int N, int K,
                              int lda, int ldb_k, int ldb_n, int ldc,
                              ll strideA, ll strideB, ll strideC,
                              int beta, int act)
{
    const int lane = threadIdx.x & 31;
    const int half = lane >> 4;       // 0: K pair {0,1}, 1: K pair {2,3}
    const int l16  = lane & 15;
    const int tm0  = blockIdx.y * MT; // first 16-tall M tile of this wave
    const int tn0  = blockIdx.x * NT; // first 16-wide N tile of this wave
    const int bz   = blockIdx.z;

    gcfp Ab = (gcfp)A  + (ll)bz * strideA;
    gcfp Bb = (gcfp)Bm + (ll)bz * strideB;
    gfp  Cb = (gfp)C   + (ll)bz * strideC;
    gcfp bi = (gcfp)bias;

    const ll   ldbk   = KS1 ? 1 : (ll)ldb_k;
    const int  k0base = 2 * half;                    // this lane's K pair offset

    // ---- per-fragment streaming pointers ----
    gcfp Ap[MT];
#pragma unroll
    for (int i = 0; i < MT; ++i) {
        int rg = (tm0 + i) * 16 + l16;               // A row supplied by this lane
        int ar = (rg < M) ? rg : (M - 1);            // clamp (guarded at store)
        Ap[i] = Ab + (ll)ar * lda + k0base;
    }
    int  colg[NT];
    gcfp Bp0[NT];                                    // element (k,   col)
    gcfp Bp1[NT];                                    // element (k+1, col)
#pragma unroll
    for (int j = 0; j < NT; ++j) {
        colg[j] = (tn0 + j) * 16 + l16;              // B col supplied by this lane
        int bc  = (colg[j] < N) ? colg[j] : (N - 1);
        Bp0[j]  = Bb + (ll)bc * ldb_n + (ll)k0base * ldbk;
        Bp1[j]  = Bp0[j] + ldbk;
    }

    // ---- accumulators (C/D layout: row = v + 8*(lane/16), col = lane%16) ---
    v8f acc[MT][NT];
#pragma unroll
    for (int i = 0; i < MT; ++i)
#pragma unroll
        for (int j = 0; j < NT; ++j) {
            acc[i][j] = (v8f){0.f, 0.f, 0.f, 0.f, 0.f, 0.f, 0.f, 0.f};
            if (beta) {
                int bc = (colg[j] < N) ? colg[j] : (N - 1);
#pragma unroll
                for (int v = 0; v < 8; ++v) {
                    int r  = (tm0 + i) * 16 + v + 8 * half;
                    int rr = (r < M) ? r : (M - 1);
                    acc[i][j][v] = Cb[(ll)rr * ldc + bc];
                }
            }
        }

    // ---- software-pipelined K loop -----------------------------------------
    v2f a_c[MT], b_c[NT];
#pragma unroll
    for (int i = 0; i < MT; ++i) {                   // prologue loads (kk = 0)
        a_c[i].x = Ap[i][0];
        a_c[i].y = Ap[i][1];
        Ap[i] += 4;
    }
#pragma unroll
    for (int j = 0; j < NT; ++j) {
        b_c[j].x = Bp0[j][0];
        b_c[j].y = Bp1[j][0];
        Bp0[j] += 4 * ldbk;
        Bp1[j] += 4 * ldbk;
    }

    for (int kk = 0; kk + 4 < K; kk += 4) {          // steady state: branch-free
        v2f a_n[MT], b_n[NT];
#pragma unroll
        for (int i = 0; i < MT; ++i) {               // next step's loads first
            a_n[i].x = Ap[i][0];
            a_n[i].y = Ap[i][1];
            Ap[i] += 4;
        }
#pragma unroll
        for (int j = 0; j < NT; ++j) {
            b_n[j].x = Bp0[j][0];
            b_n[j].y = Bp1[j][0];
            Bp0[j] += 4 * ldbk;
            Bp1[j] += 4 * ldbk;
        }
#pragma unroll
        for (int i = 0; i < MT; ++i)
#pragma unroll
            for (int j = 0; j < NT; ++j)
                acc[i][j] = __builtin_amdgcn_wmma_f32_16x16x4_f32(
                                false, a_c[i], false, b_c[j], (short)0,
                                acc[i][j], false, false);
#pragma unroll
        for (int i = 0; i < MT; ++i) a_c[i] = a_n[i];
#pragma unroll
        for (int j = 0; j < NT; ++j) b_c[j] = b_n[j];
    }
#pragma unroll
    for (int i = 0; i < MT; ++i)                     // peeled final burst
#pragma unroll
        for (int j = 0; j < NT; ++j)
            acc[i][j] = __builtin_amdgcn_wmma_f32_16x16x4_f32(
                            false, a_c[i], false, b_c[j], (short)0,
                            acc[i][j], false, false);

    // ---- epilogue: bias + activation + guarded store -----------------------
#pragma unroll
    for (int j = 0; j < NT; ++j) {
        int bc = (colg[j] < N) ? colg[j] : (N - 1);
        float bv = bias ? bi[bc] : 0.0f;
#pragma unroll
        for (int i = 0; i < MT; ++i)
#pragma unroll
            for (int v = 0; v < 8; ++v) {
                int r = (tm0 + i) * 16 + v + 8 * half;
                if (r < M && colg[j] < N) {
                    float val = acc[i][j][v] + bv;
                    if (act == 1)      val = tanhf(val);
                    else if (act == 2) val = expf(val);
                    Cb[(ll)r * ldc + colg[j]] = val;
                }
            }
    }
}

#define GNT 4   // N tiles per wave
static inline dim3 gemm_grid(int M, int N, int batch, int mt) {
    return dim3((N + 16 * GNT - 1) / (16 * GNT),
                (M + 16 * mt - 1) / (16 * mt), batch);
}

// ---------------- embedding gather: embed[b,t,:] = embedding[dec_x[b,t],:] ---
__global__ void embed_kernel(const int* __restrict__ dec_x,
                             const float* __restrict__ embedding,
                             float* __restrict__ embed)
{
    int idx = blockIdx.x * blockDim.x + threadIdx.x;
    if (idx >= Bz_ * T_ * EMB_) return;
    int bt = idx / EMB_;
    int e  = idx - bt * EMB_;
    embed[idx] = embedding[(ll)dec_x[bt] * EMB_ + e];
}

// ---------------- init h,c from enc_h, enc_c --------------------------------
__global__ void init_state_kernel(float* __restrict__ h, float* __restrict__ c,
                                  const float* __restrict__ enc_h,
                                  const float* __restrict__ enc_c)
{
    int i = blockIdx.x * blockDim.x + threadIdx.x;
    if (i >= Bz_ * H_) return;
    h[i] = enc_h[i];
    c[i] = enc_c[i];
}

// ---------------- LSTM gates: consume z[B,4H], update h,c; record hidden ----
__global__ void lstm_gate_kernel(const float* __restrict__ z,
                                 float* __restrict__ h, float* __restrict__ c,
                                 float* __restrict__ hidden, int t)
{
    int idx = blockIdx.x * blockDim.x + threadIdx.x;
    if (idx >= Bz_ * H_) return;
    int b = idx / H_;
    int j = idx - b * H_;
    const float* zr = z + (ll)b * 4 * H_;
    float zi = zr[j];
    float zf = zr[H_ + j];
    float zg = zr[2 * H_ + j];
    float zo = zr[3 * H_ + j];
    float si = 1.0f / (1.0f + expf(-zi));
    float sf = 1.0f / (1.0f + expf(-zf));
    float so = 1.0f / (1.0f + expf(-zo));
    float cn = sf * c[idx] + si * tanhf(zg);
    float hn = so * tanhf(cn);
    c[idx] = cn;
    h[idx] = hn;
    hidden[((ll)b * T_ + t) * H_ + j] = hn;
}

// ---------------- masked softmax over S, in place (one block per (b,t)) -----
__global__ void softmax_mask_kernel(float* __restrict__ scores,
                                    const int* __restrict__ enc_len)
{
    __shared__ float red[S_];
    int bt = blockIdx.x;
    int b  = bt / T_;
    int s  = threadIdx.x;
    float v = scores[(ll)bt * S_ + s];
    if (s >= enc_len[b]) v = NEG_;
    red[s] = v;
    __syncthreads();
    for (int off = S_ / 2; off > 0; off >>= 1) {
        if (s < off) red[s] = fmaxf(red[s], red[s + off]);
        __syncthreads();
    }
    float m = red[0];
    __syncthreads();
    float e = expf(v - m);
    red[s] = e;
    __syncthreads();
    for (int off = S_ / 2; off > 0; off >>= 1) {
        if (s < off) red[s] += red[s + off];
        __syncthreads();
    }
    scores[(ll)bt * S_ + s] = e / red[0];
}

// ---------------- cs = exp(where(mask, cs, NEG)) ----------------------------
__global__ void mask_exp_kernel(float* __restrict__ cs,
                                const int* __restrict__ enc_len)
{
    int idx = blockIdx.x * blockDim.x + threadIdx.x;
    if (idx >= Bz_ * T_ * S_) return;
    int bt = idx / S_;
    int s  = idx - bt * S_;
    int b  = bt / T_;
    float v = cs[idx];
    if (s >= enc_len[b]) v = NEG_;
    cs[idx] = expf(v);          // exp(NEG) == 0
}

// ---------------- pad OOV columns with 1e-10 --------------------------------
__global__ void fill_oov_kernel(float* __restrict__ out)
{
    int idx = blockIdx.x * blockDim.x + threadIdx.x;
    if (idx >= Bz_ * T_ * OOV_) return;
    int bt = idx / OOV_;
    int o  = idx - bt * OOV_;
    out[(ll)bt * VT_ + VOCAB_ + o] = 1e-10f;
}

// ---------------- scatter-add copy scores into extended vocab ---------------
__global__ void scatter_kernel(float* __restrict__ out,
                               const float* __restrict__ cs,
                               const int* __restrict__ enc_x)
{
    int idx = blockIdx.x * blockDim.x + threadIdx.x;
    if (idx >= Bz_ * T_ * S_) return;
    int bt = idx / S_;
    int s  = idx - bt * S_;
    int b  = bt / T_;
    int col = enc_x[(ll)b * S_ + s];
    atomicAdd(&out[(ll)bt * VT_ + col], cs[idx]);
}

// ---------------- row log-normalize over VT (one block per (b,t)) -----------
__global__ void normalize_kernel(float* __restrict__ out)
{
    __shared__ float red[256];
    int bt = blockIdx.x;
    float* row = out + (ll)bt * VT_;
    int tid = threadIdx.x;
    float acc = 0.0f;
    for (int i = tid; i < VT_; i += 256) acc += row[i];
    red[tid] = acc;
    __syncthreads();
    for (int off = 128; off > 0; off >>= 1) {
        if (tid < off) red[tid] += red[tid + off];
        __syncthreads();
    }
    float sum = red[0];
    for (int i = tid; i < VT_; i += 256) row[i] = logf(row[i] / sum);
}

// ---------------- copy final (h, c) into output tail ------------------------
__global__ void copy_state_kernel(float* __restrict__ out_tail,
                                  const float* __restrict__ h,
                                  const float* __restrict__ c)
{
    int i = blockIdx.x * blockDim.x + threadIdx.x;
    if (i >= Bz_ * H_) return;
    out_tail[i]            = h[i];
    out_tail[Bz_ * H_ + i] = c[i];
}

// ---------------------------------------------------------------------------
extern "C" void kernel_launch(void* const* d_in, const int* in_sizes, int n_in,
                              void* d_out, int out_size, void* d_ws, size_t ws_size,
                              hipStream_t stream)
{
    const int*   dec_x      = (const int*)  d_in[0];
    const int*   enc_x_oov  = (const int*)  d_in[1];
    const int*   enc_len    = (const int*)  d_in[2];
    const float* enc_output = (const float*)d_in[3];
    const float* enc_h      = (const float*)d_in[4];
    const float* enc_c      = (const float*)d_in[5];
    const float* embedding  = (const float*)d_in[8];
    const float* lstm_k     = (const float*)d_in[9];
    const float* lstm_r     = (const float*)d_in[10];
    const float* lstm_b     = (const float*)d_in[11];
    const float* attn_w     = (const float*)d_in[12];
    const float* out_w      = (const float*)d_in[13];
    const float* out_b      = (const float*)d_in[14];
    const float* gen_w      = (const float*)d_in[15];
    const float* copy_w     = (const float*)d_in[16];
    const float* copy_b     = (const float*)d_in[17];

    float* out = (float*)d_out;

    // ---- workspace layout (floats) ----
    float* ws       = (float*)d_ws;
    float* embed    = ws;                                  // B*T*EMB   = 65536
    float* hidden   = embed   + (ll)Bz_ * T_ * EMB_;       // B*T*H     = 131072
    float* hbuf     = hidden  + (ll)Bz_ * T_ * H_;         // B*H       = 16384
    float* cbuf     = hbuf    + (ll)Bz_ * H_;              // B*H       = 16384
    float* zbuf     = cbuf    + (ll)Bz_ * H_;              // B*4H      = 65536
    float* qbuf     = zbuf    + (ll)Bz_ * 4 * H_;          // B*T*ENC   = 131072
    float* scores   = qbuf    + (ll)Bz_ * T_ * ENC_;       // B*T*S     = 65536
    float* context  = scores  + (ll)Bz_ * T_ * S_;         // B*T*ENC   = 131072
    float* attn_out = context + (ll)Bz_ * T_ * ENC_;       // B*T*H     = 131072
    float* kbuf     = attn_out+ (ll)Bz_ * T_ * H_;         // B*S*H     = 4194304
    float* csbuf    = kbuf    + (ll)Bz_ * S_ * H_;         // B*T*S     = 65536

    // ---- embedding lookup + initial state ----
    embed_kernel<<<(Bz_*T_*EMB_ + 255)/256, 256, 0, stream>>>(dec_x, embedding, embed);
    init_state_kernel<<<(Bz_*H_ + 255)/256, 256, 0, stream>>>(hbuf, cbuf, enc_h, enc_c);

    // ---- LSTM over T steps: z = x_t@Wk + h@Wr + b ; gates ----
    for (int t = 0; t < T_; ++t) {
        // z = embed[:,t,:] @ lstm_k                (M=32, N=2048, K=256)
        wmma_gemm_f32<2, GNT, false><<<gemm_grid(Bz_, 4*H_, 1, 2), 32, 0, stream>>>(
            embed + (ll)t * EMB_, lstm_k, zbuf, nullptr,
            Bz_, 4*H_, EMB_,
            T_*EMB_, 4*H_, 1, 4*H_, 0, 0, 0, /*beta=*/0, /*act=*/0);
        // z += h @ lstm_r + lstm_b                 (M=32, N=2048, K=512)
        wmma_gemm_f32<2, GNT, false><<<gemm_grid(Bz_, 4*H_, 1, 2), 32, 0, stream>>>(
            hbuf, lstm_r, zbuf, lstm_b,
            Bz_, 4*H_, H_,
            H_, 4*H_, 1, 4*H_, 0, 0, 0, /*beta=*/1, /*act=*/0);
        lstm_gate_kernel<<<(Bz_*H_ + 255)/256, 256, 0, stream>>>(
            zbuf, hbuf, cbuf, hidden, t);
    }

    // ---- q = hidden @ attn_w                    (M=256, N=512, K=512) ----
    wmma_gemm_f32<2, GNT, false><<<gemm_grid(Bz_*T_, ENC_, 1, 2), 32, 0, stream>>>(
        hidden, attn_w, qbuf, nullptr,
        Bz_*T_, ENC_, H_,
        H_, ENC_, 1, ENC_, 0, 0, 0, 0, 0);

    // ---- scores[b] = q[b] @ enc_output[b]^T     (batched, M=8, N=256, K=512)
    wmma_gemm_f32<1, GNT, true><<<gemm_grid(T_, S_, Bz_, 1), 32, 0, stream>>>(
        qbuf, enc_output, scores, nullptr,
        T_, S_, ENC_,
        ENC_, /*ldb_k=*/1, /*ldb_n=*/ENC_, S_,
        (ll)T_*ENC_, (ll)S_*ENC_, (ll)T_*S_, 0, 0);

    softmax_mask_kernel<<<Bz_*T_, S_, 0, stream>>>(scores, enc_len);

    // ---- context[b] = attn[b] @ enc_output[b]   (batched, M=8, N=512, K=256)
    wmma_gemm_f32<1, GNT, false><<<gemm_grid(T_, ENC_, Bz_, 1), 32, 0, stream>>>(
        scores, enc_output, context, nullptr,
        T_, ENC_, S_,
        S_, ENC_, 1, ENC_,
        (ll)T_*S_, (ll)S_*ENC_, (ll)T_*ENC_, 0, 0);

    // ---- attn_out = tanh(context@out_w[0:512] + hidden@out_w[512:1024] + b)
    wmma_gemm_f32<2, GNT, false><<<gemm_grid(Bz_*T_, H_, 1, 2), 32, 0, stream>>>(
        context, out_w, attn_out, nullptr,
        Bz_*T_, H_, ENC_,
        ENC_, H_, 1, H_, 0, 0, 0, /*beta=*/0, /*act=*/0);
    wmma_gemm_f32<2, GNT, false><<<gemm_grid(Bz_*T_, H_, 1, 2), 32, 0, stream>>>(
        hidden, out_w + (ll)ENC_ * H_, attn_out, out_b,
        Bz_*T_, H_, H_,
        H_, H_, 1, H_, 0, 0, 0, /*beta=*/1, /*act=*/1);

    // ---- gen = exp(attn_out @ gen_w) written straight into out (ldc = VT) --
    wmma_gemm_f32<2, GNT, false><<<gemm_grid(Bz_*T_, VOCAB_, 1, 2), 32, 0, stream>>>(
        attn_out, gen_w, out, nullptr,
        Bz_*T_, VOCAB_, H_,
        H_, VOCAB_, 1, VT_, 0, 0, 0, /*beta=*/0, /*act=*/2);

    fill_oov_kernel<<<(Bz_*T_*OOV_ + 255)/256, 256, 0, stream>>>(out);

    // ---- k = tanh(enc_output @ copy_w + copy_b) (M=8192, N=512, K=512) ----
    wmma_gemm_f32<2, GNT, false><<<gemm_grid(Bz_*S_, H_, 1, 2), 32, 0, stream>>>(
        enc_output, copy_w, kbuf, copy_b,
        Bz_*S_, H_, ENC_,
        ENC_, H_, 1, H_, 0, 0, 0, /*beta=*/0, /*act=*/1);

    // ---- cs[b] = attn_out[b] @ k[b]^T           (batched, M=8, N=256, K=512)
    wmma_gemm_f32<1, GNT, true><<<gemm_grid(T_, S_, Bz_, 1), 32, 0, stream>>>(
        attn_out, kbuf, csbuf, nullptr,
        T_, S_, H_,
        H_, /*ldb_k=*/1, /*ldb_n=*/H_, S_,
        (ll)T_*H_, (ll)S_*H_, (ll)T_*S_, 0, 0);

    mask_exp_kernel<<<(Bz_*T_*S_ + 255)/256, 256, 0, stream>>>(csbuf, enc_len);
    scatter_kernel<<<(Bz_*T_*S_ + 255)/256, 256, 0, stream>>>(out, csbuf, enc_x_oov);

    normalize_kernel<<<Bz_*T_, 256, 0, stream>>>(out);

    copy_state_kernel<<<(Bz_*H_ + 255)/256, 256, 0, stream>>>(
        out + (ll)Bz_ * T_ * VT_, hbuf, cbuf);

    (void)in_sizes; (void)n_in; (void)out_size; (void)ws_size;
}